// PointConv_SM_8323646619716
// MI455X (gfx1250) — compile-verified
//
#include <hip/hip_runtime.h>

typedef __attribute__((ext_vector_type(2))) float v2f;
typedef __attribute__((ext_vector_type(4))) float v4f;
typedef __attribute__((ext_vector_type(8))) float v8f;
typedef __attribute__((ext_vector_type(4))) unsigned int v4u;
typedef __attribute__((ext_vector_type(4))) int v4i;
typedef __attribute__((ext_vector_type(8))) int v8i;

namespace {
constexpr int NPTS  = 65536;
constexpr int KNN   = 16;
constexpr int CIN   = 64;
constexpr int COUT  = 64;
constexpr int WCOLS = 67;            // w_1x1 columns (64 fea + 3 rel)
constexpr int WTOT  = COUT * WCOLS;  // 4288 floats
constexpr int P     = 128;           // points per block
constexpr int CS    = 76;            // s[p][c] row stride: bank-conflict-free
constexpr int NTHREADS = 256;
}

// ---------- pass 1: fea (64,N) -> feaT (N,64), enables b128 channel gathers ----------
__global__ __launch_bounds__(256) void fea_transpose(const float* __restrict__ fea,
                                                     float* __restrict__ feaT) {
  __shared__ float t[64 * 65];                 // 16640 B, padded stride
  const int tid = threadIdx.x;
  const int n0  = blockIdx.x * 64;
  #pragma unroll
  for (int i = 0; i < 16; i++) {               // coalesced reads along n
    int idx = tid + (i << 8);
    int c = idx >> 6, p = idx & 63;
    t[c * 65 + p] = fea[(size_t)c * NPTS + n0 + p];
  }
  __syncthreads();
  #pragma unroll
  for (int i = 0; i < 4; i++) {                // b128 coalesced writes along c
    int idx = tid + (i << 8);                  // 0..1023
    int p = idx >> 4, g = idx & 15;
    v4f v;
    v.x = t[(4 * g + 0) * 65 + p];
    v.y = t[(4 * g + 1) * 65 + p];
    v.z = t[(4 * g + 2) * 65 + p];
    v.w = t[(4 * g + 3) * 65 + p];
    ((v4f*)feaT)[(size_t)(n0 + p) * 16 + g] = v;
  }
}

// ---------- pass 2: gather-sum + f32 WMMA + depthwise scale ----------
template <bool USET>
__global__ __launch_bounds__(NTHREADS) void pointconv_wmma_f32(
    const float* __restrict__ sample,   // (N,3)
    const float* __restrict__ rel,      // (3,16,N)
    const float* __restrict__ fea,      // (64,N)
    const float* __restrict__ feaT,     // (N,64)  [USET only]
    const int*   __restrict__ knn,      // (N,16)
    const float* __restrict__ w,        // (64,67)
    const float* __restrict__ bias,     // (64)
    const float* __restrict__ cdw,      // (64,5,5,5)
    float* __restrict__ out)            // (64,N)
{
  __shared__ float s_s[P * CS];         // 38912 B : s[p][c] channel sums (c 0..67)
  __shared__ float s_w[WTOT];           // 17152 B : packed weights (64 x 67)
  __shared__ int   s_idx[P * 15];       //  7680 B : neighbor indices j=1..15
  __shared__ int   s_off[P];            //   512 B : conv_dw offsets

  const int tid = threadIdx.x;
  const int n0  = blockIdx.x * P;

  // ---- stage weights into LDS via the Tensor Data Mover (wave 0 only) ----
#if __has_builtin(__builtin_amdgcn_tensor_load_to_lds) && __has_builtin(__builtin_amdgcn_s_wait_tensorcnt)
  if (tid < 32) {
    unsigned long long ga = (unsigned long long)(uintptr_t)w;
    unsigned int lds = (unsigned int)(uintptr_t)(&s_w[0]);
    v4u g0;
    g0.x = 1u;                                       // count=1 user descriptor
    g0.y = lds;                                      // lds_addr (bytes)
    g0.z = (unsigned int)(ga & 0xffffffffull);       // global_addr[31:0]
    g0.w = (unsigned int)(ga >> 32) | 0x80000000u;   // global_addr[56:32] | type=2
    v8i g1;
    g1[0] = 0x20000;                                 // data_size=2 (4B); no mask/pad
    g1[1] = (WTOT & 0xffff) << 16;                   // tensor_dim0[15:0]
    g1[2] = (WTOT >> 16) | (1 << 16);                // tensor_dim0[31:16] | tensor_dim1=1
    g1[3] = (WTOT & 0xffff) << 16;                   // tile_dim0 = 4288
    g1[4] = 1;                                       // tile_dim1 = 1
    g1[5] = WTOT;                                    // tensor_dim0_stride[31:0]
    g1[6] = 0;
    g1[7] = 0;
    v4i z4 = {0, 0, 0, 0};
#if __clang_major__ >= 23
    v8i z8 = {0, 0, 0, 0, 0, 0, 0, 0};
    __builtin_amdgcn_tensor_load_to_lds(g0, g1, z4, z4, z8, 0);
#else
    __builtin_amdgcn_tensor_load_to_lds(g0, g1, z4, z4, 0);
#endif
    __builtin_amdgcn_s_wait_tensorcnt(0);            // s_wait_tensorcnt 0
  }
#else
  for (int i = tid; i < WTOT; i += NTHREADS) s_w[i] = w[i];
#endif

  // ---- stage neighbor indices (drop j==0: center handled analytically) ----
  for (int i = tid; i < P * KNN; i += NTHREADS) {
    int p = i >> 4, j = i & 15;
    int v = knn[(size_t)(n0 + p) * KNN + j];
    if (j > 0) s_idx[p * 15 + (j - 1)] = v;
  }
  // ---- per point: rel-xyz channel sums + depthwise-kernel offset ----
  for (int p = tid; p < P; p += NTHREADS) {
    int n = n0 + p;
    float r0 = 0.f, r1 = 0.f, r2 = 0.f;
    #pragma unroll
    for (int kk = 0; kk < KNN; kk++) {
      r0 += rel[(size_t)(0 * KNN + kk) * NPTS + n];
      r1 += rel[(size_t)(1 * KNN + kk) * NPTS + n];
      r2 += rel[(size_t)(2 * KNN + kk) * NPTS + n];
    }
    v4f rv = {r0, r1, r2, 0.f};                      // c=67 stays zero (K pad)
    *(v4f*)&s_s[p * CS + 64] = rv;
    float sx = sample[3 * n + 0];
    float sy = sample[3 * n + 1];
    float sz = sample[3 * n + 2];
    int cx = (int)((fminf(fmaxf(sx, -0.99999f), 0.99999f) * 5.f) / 2.f) + 2;
    int cy = (int)((fminf(fmaxf(sy, -0.99999f), 0.99999f) * 5.f) / 2.f) + 2;
    int cz = (int)((fminf(fmaxf(sz, -0.99999f), 0.99999f) * 5.f) / 2.f) + 2;
    s_off[p] = cz * 25 + cy * 5 + cx;
  }
  __syncthreads();

  // ---- gather-sum: s[p][c] = sum_j fea[c, idx[p,j]] - 14*fea[c, n0+p] ----
  if (USET) {
    // half-wave (16 lanes) per point: each neighbor row = 16 contiguous b128
    // loads (256 B fully consumed), accumulate 4 channels per lane.
    const int l  = tid & 15;
    const int hw = tid >> 4;                         // 16 half-waves
    const v4f* bp = (const v4f*)feaT;
    #pragma unroll 1
    for (int it = 0; it < P / 16; it++) {            // 8 points per half-wave
      int p = hw + (it << 4);
      v4f acc = bp[(size_t)(n0 + p) * 16 + l] * (-14.f);
      #pragma unroll
      for (int j = 0; j < 15; j++) {
        int m = s_idx[p * 15 + j];                   // LDS broadcast
        acc += bp[(size_t)m * 16 + l];
      }
      *(v4f*)&s_s[p * CS + 4 * l] = acc;             // ds_store_b128, bank-perfect
    }
  } else {
    // fallback (no workspace): scalar gathers straight from fea
    #pragma unroll 1
    for (int i = 0; i < (CIN * P) / NTHREADS; i++) { // 32 iters/thread
      int pair = tid + (i << 8);
      int c = pair >> 7;
      int p = pair & (P - 1);
      const float* frow = fea + (size_t)c * NPTS;
      float acc = -14.f * frow[n0 + p];
      #pragma unroll
      for (int j = 0; j < 15; j++) acc += frow[s_idx[p * 15 + j]];
      s_s[p * CS + c] = acc;
    }
  }
  __syncthreads();

  // ---- WMMA: pre[o,p] = sum_c w[o,c] * s[c,p], K = 68 = 17 x 4 (pure f32) ----
  const int lane = tid & 31;
  const int wv   = tid >> 5;       // 8 waves = 8 point-groups of 16
  const int half = lane >> 4;
  const int pl   = lane & 15;
  const int p    = wv * 16 + pl;

  // B fragments (4x16 f32): vgpr r holds K = 4*kk + 2*half + r; b64 LDS reads,
  // loaded once and reused across all 4 output-channel groups.
  v2f bf[17];
  #pragma unroll
  for (int kk = 0; kk < 17; kk++) {
    int c = kk * 4 + 2 * half;
    bf[kk] = *(const v2f*)&s_s[p * CS + c];
  }
  const int poff = s_off[p];
  const int n = n0 + p;

  #pragma unroll 1
  for (int og = 0; og < 4; og++) {
    v8f acc = {};
    #pragma unroll
    for (int kk = 0; kk < 17; kk++) {
      int c = kk * 4 + 2 * half;
      int o = og * 16 + pl;                          // A: M = lane%16
      v2f af;
      af.x = s_w[o * WCOLS + c];
      af.y = (c + 1 < WCOLS) ? s_w[o * WCOLS + c + 1] : 0.f;  // K-pad col 67
      acc = __builtin_amdgcn_wmma_f32_16x16x4_f32(
          false, af, false, bf[kk], (short)0, acc, false, false);
    }
    // D layout: vgpr v -> M = v + 8*half ; N = lane%16
    #pragma unroll
    for (int v = 0; v < 8; v++) {
      int o = og * 16 + v + 8 * half;
      float kdw = cdw[o * 125 + poff];
      float bo  = bias[o];
      out[(size_t)o * NPTS + n] = kdw * (acc[v] + 16.f * bo);
    }
  }
}

extern "C" void kernel_launch(void* const* d_in, const int* in_sizes, int n_in,
                              void* d_out, int out_size, void* d_ws, size_t ws_size,
                              hipStream_t stream) {
  (void)in_sizes; (void)n_in; (void)out_size;
  const float* sample = (const float*)d_in[0];
  const float* rel    = (const float*)d_in[1];
  const float* fea    = (const float*)d_in[2];
  const int*   knn    = (const int*)  d_in[3];
  const float* w      = (const float*)d_in[4];
  const float* bias   = (const float*)d_in[5];
  const float* cdw    = (const float*)d_in[6];
  float* out = (float*)d_out;

  const size_t feat_bytes = (size_t)CIN * NPTS * sizeof(float);
  const bool useT = (d_ws != nullptr) && (ws_size >= feat_bytes);

  if (useT) {
    float* feaT = (float*)d_ws;
    fea_transpose<<<dim3(NPTS / 64), dim3(256), 0, stream>>>(fea, feaT);
    pointconv_wmma_f32<true><<<dim3(NPTS / P), dim3(NTHREADS), 0, stream>>>(
        sample, rel, fea, feaT, knn, w, bias, cdw, out);
  } else {
    pointconv_wmma_f32<false><<<dim3(NPTS / P), dim3(NTHREADS), 0, stream>>>(
        sample, rel, fea, fea, knn, w, bias, cdw, out);
  }
}